// IsomporphismOneHotConv_73890617360770
// MI455X (gfx1250) — compile-verified
//
#include <hip/hip_runtime.h>

#define NN   2048
#define EE   32768
#define CIN  128
#define COUT 128
#define OHCH 8
#define KIN1 136          // CIN + OHCH
#define MAXDEG 96         // Poisson(16) over 2048 bins: P(deg > 50) ~ 0
#define BN_EPS 1e-5f

typedef __attribute__((ext_vector_type(2))) float v2f;
typedef __attribute__((ext_vector_type(8))) float v8f;

// ---- build per-receiver sorted send-lists (deterministic edge order) ----
__global__ __launch_bounds__(256)
void k_build_lists(const int* __restrict__ adj, int* __restrict__ deg,
                   int* __restrict__ lists) {
  const int n = blockIdx.x;
  const int tid = threadIdx.x;
  __shared__ int cnt;
  __shared__ int ls[MAXDEG];
  if (tid == 0) cnt = 0;
  __syncthreads();
  for (int e = tid; e < EE; e += blockDim.x) {
    if (adj[EE + e] == n) {
      int p = atomicAdd(&cnt, 1);
      if (p < MAXDEG) ls[p] = adj[e];
    }
  }
  __syncthreads();
  int d = cnt < MAXDEG ? cnt : MAXDEG;
  if (tid == 0) {
    // insertion sort -> canonical order regardless of atomic arrival order
    for (int i = 1; i < d; ++i) {
      int key = ls[i], j = i - 1;
      while (j >= 0 && ls[j] > key) { ls[j + 1] = ls[j]; --j; }
      ls[j + 1] = key;
    }
    deg[n] = d;
  }
  __syncthreads();
  for (int i = tid; i < d; i += blockDim.x) lists[n * MAXDEG + i] = ls[i];
}

// ---- agg_x[n] = sum over sorted in-edges of x[send] (no atomics) ----
__global__ __launch_bounds__(128)
void k_gather_x(const float* __restrict__ x, const int* __restrict__ deg,
                const int* __restrict__ lists, float* __restrict__ agg_x) {
  const int n = blockIdx.x;
  const int c = threadIdx.x;                 // 128 channels
  const int d = deg[n];
  float acc = 0.f;
  for (int i = 0; i < d; ++i) {
    int s = lists[n * MAXDEG + i];           // uniform per block -> scalar load
    acc += x[(size_t)s * CIN + c];
  }
  agg_x[(size_t)n * CIN + c] = acc;
}

// ---- new_oh[n] = onehot[n] + sum over sorted in-edges of onehot[send] ----
__global__ __launch_bounds__(256)
void k_gather_oh(const float* __restrict__ oh, const int* __restrict__ deg,
                 const int* __restrict__ lists, float* __restrict__ out_oh) {
  const int n = blockIdx.x;
  const int tid = threadIdx.x;
  const int d = deg[n];
  __shared__ int ls[MAXDEG];
  for (int i = tid; i < d; i += blockDim.x) ls[i] = lists[n * MAXDEG + i];
  __syncthreads();
  const int n4 = NN / 4;                     // 512 float4 per row
  const float4* ohv = (const float4*)oh;
  float4* outv = (float4*)out_oh;
  for (int col = tid; col < n4; col += blockDim.x) {
    float4 acc = ohv[(size_t)n * n4 + col];
    for (int i = 0; i < d; ++i) {
      if (i + 1 < d)
        __builtin_prefetch(&ohv[(size_t)ls[i + 1] * n4 + col], 0, 3);
      float4 v = ohv[(size_t)ls[i] * n4 + col];
      acc.x += v.x; acc.y += v.y; acc.z += v.z; acc.w += v.w;
    }
    outv[(size_t)n * n4 + col] = acc;
  }
}

// ---- per-row bitonic sort in LDS + fused conv pipe + readout ----
__global__ __launch_bounds__(1024)
void k_sort_conv(const float* __restrict__ new_oh,
                 const float* __restrict__ cw1, const float* __restrict__ cb1,
                 const float* __restrict__ cw2, const float* __restrict__ cb2,
                 const float* __restrict__ Wr,  const float* __restrict__ br,
                 float* __restrict__ readout) {
  __shared__ float s[NN];                     // 8 KB sorted row
  __shared__ float wc1[24], wb1[8], wc2[384], wb2[16];
  __shared__ float part[32][16];              // per-wave partials
  __shared__ float m16[16];
  const int tid = threadIdx.x;
  const int row = blockIdx.x;

  for (int i = tid; i < NN; i += blockDim.x) s[i] = new_oh[(size_t)row * NN + i];
  if (tid < 24) wc1[tid] = cw1[tid];
  if (tid < 8)  wb1[tid] = cb1[tid];
  for (int i = tid; i < 384; i += blockDim.x) wc2[i] = cw2[i];
  if (tid < 16) wb2[tid] = cb2[tid];
  __syncthreads();

  // bitonic sort ascending, n = 2048
  for (int k = 2; k <= NN; k <<= 1) {
    for (int j = k >> 1; j > 0; j >>= 1) {
      for (int i = tid; i < NN; i += blockDim.x) {
        int ixj = i ^ j;
        if (ixj > i) {
          float a = s[i], b = s[ixj];
          if ((a > b) == ((i & k) == 0)) { s[i] = b; s[ixj] = a; }
        }
      }
      __syncthreads();
    }
  }

  // conv1d(1->8,k3,p1) -> relu -> conv1d(8->16,k3,p1) -> relu -> mean over L
  float acc[16];
#pragma unroll
  for (int c = 0; c < 16; ++c) acc[c] = 0.f;

  for (int l = tid; l < NN; l += blockDim.x) {
    float sv[5];
#pragma unroll
    for (int dd = 0; dd < 5; ++dd) {
      int q = l - 2 + dd;
      sv[dd] = (q >= 0 && q < NN) ? s[q] : 0.f;
    }
    float h1[3][8];
#pragma unroll
    for (int p = 0; p < 3; ++p) {
      int pos = l - 1 + p;
      bool valid = (pos >= 0 && pos < NN);
#pragma unroll
      for (int c1 = 0; c1 < 8; ++c1) {
        float t = wb1[c1] + wc1[c1 * 3 + 0] * sv[p]
                          + wc1[c1 * 3 + 1] * sv[p + 1]
                          + wc1[c1 * 3 + 2] * sv[p + 2];
        h1[p][c1] = valid ? fmaxf(t, 0.f) : 0.f;
      }
    }
#pragma unroll
    for (int c2 = 0; c2 < 16; ++c2) {
      float t = wb2[c2];
#pragma unroll
      for (int c1 = 0; c1 < 8; ++c1) {
        t += wc2[c2 * 24 + c1 * 3 + 0] * h1[0][c1]
           + wc2[c2 * 24 + c1 * 3 + 1] * h1[1][c1]
           + wc2[c2 * 24 + c1 * 3 + 2] * h1[2][c1];
      }
      acc[c2] += fmaxf(t, 0.f);
    }
  }

  // deterministic reduction: wave32 shuffle tree, then fixed-order cross-wave sum
#pragma unroll
  for (int c = 0; c < 16; ++c)
#pragma unroll
    for (int off = 16; off > 0; off >>= 1)
      acc[c] += __shfl_xor(acc[c], off, 32);
  const int wv = tid >> 5, ln = tid & 31;
  if (ln == 0) {
#pragma unroll
    for (int c = 0; c < 16; ++c) part[wv][c] = acc[c];
  }
  __syncthreads();
  if (tid < 16) {
    float t = 0.f;
    for (int w = 0; w < 32; ++w) t += part[w][tid];
    m16[tid] = t;
  }
  __syncthreads();

  if (tid < OHCH) {
    float t = br[tid];
#pragma unroll
    for (int c = 0; c < 16; ++c)
      t += (m16[c] * (1.f / NN)) * Wr[c * OHCH + tid];
    readout[(size_t)row * OHCH + tid] = t;
  }
}

// ---- GEMM1: concat(agg_x, readout)[N,136] @ W1[136,128] + b1 ----
__global__ __launch_bounds__(256)
void k_gemm1(const float* __restrict__ agg_x, const float* __restrict__ readout,
             const float* __restrict__ W1, const float* __restrict__ b1,
             float* __restrict__ x2pre) {
  const int wave = threadIdx.x >> 5;          // 8 waves -> 8 column tiles
  const int lane = threadIdx.x & 31;
  const int half = lane >> 4;
  const int lm   = lane & 15;
  const int m0 = blockIdx.x * 16;
  const int n0 = wave * 16;
  const int rowA = m0 + lm;
  const float* fx = agg_x   + (size_t)rowA * CIN;
  const float* fr = readout + (size_t)rowA * OHCH;

  v8f c = {};
  for (int k = 0; k < CIN; k += 4) {          // main K from agg_x
    int k0 = k + 2 * half;
    v2f a = { fx[k0], fx[k0 + 1] };
    v2f b = { W1[(size_t)k0 * COUT + n0 + lm],
              W1[(size_t)(k0 + 1) * COUT + n0 + lm] };
    c = __builtin_amdgcn_wmma_f32_16x16x4_f32(false, a, false, b, (short)0, c, false, false);
  }
  for (int k = CIN; k < KIN1; k += 4) {       // readout tail (8 channels)
    int k0 = k + 2 * half;
    v2f a = { fr[k0 - CIN], fr[k0 + 1 - CIN] };
    v2f b = { W1[(size_t)k0 * COUT + n0 + lm],
              W1[(size_t)(k0 + 1) * COUT + n0 + lm] };
    c = __builtin_amdgcn_wmma_f32_16x16x4_f32(false, a, false, b, (short)0, c, false, false);
  }
  float bias = b1[n0 + lm];
#pragma unroll
  for (int r = 0; r < 8; ++r) {
    int m = m0 + r + 8 * half;
    x2pre[(size_t)m * COUT + n0 + lm] = c[r] + bias;
  }
}

// ---- BatchNorm batch statistics (deterministic tree reduction) ----
__global__ __launch_bounds__(256)
void k_bnstats(const float* __restrict__ x2pre, float* __restrict__ mu,
               float* __restrict__ rstd) {
  const int ch = blockIdx.x;
  const int tid = threadIdx.x;
  float sum = 0.f, sq = 0.f;
  for (int r = tid; r < NN; r += blockDim.x) {
    float v = x2pre[(size_t)r * COUT + ch];
    sum += v; sq += v * v;
  }
  __shared__ float ssum[256], ssq[256];
  ssum[tid] = sum; ssq[tid] = sq;
  __syncthreads();
  for (int off = 128; off > 0; off >>= 1) {
    if (tid < off) { ssum[tid] += ssum[tid + off]; ssq[tid] += ssq[tid + off]; }
    __syncthreads();
  }
  if (tid == 0) {
    float m = ssum[0] * (1.f / NN);
    float v = ssq[0] * (1.f / NN) - m * m;
    mu[ch] = m;
    rstd[ch] = rsqrtf(v + BN_EPS);
  }
}

// ---- GEMM2: relu(bn(x2pre)) @ W2[128,128] + b2 -> d_out ----
__global__ __launch_bounds__(256)
void k_gemm2(const float* __restrict__ x2pre,
             const float* __restrict__ mu, const float* __restrict__ rstd,
             const float* __restrict__ gamma, const float* __restrict__ beta,
             const float* __restrict__ W2, const float* __restrict__ b2,
             float* __restrict__ out) {
  const int wave = threadIdx.x >> 5;
  const int lane = threadIdx.x & 31;
  const int half = lane >> 4;
  const int lm   = lane & 15;
  const int m0 = blockIdx.x * 16;
  const int n0 = wave * 16;
  const int rowA = m0 + lm;
  const float* xr = x2pre + (size_t)rowA * COUT;

  v8f c = {};
  for (int k = 0; k < COUT; k += 4) {
    int k0 = k + 2 * half;
    int k1 = k0 + 1;
    float a0 = fmaxf((xr[k0] - mu[k0]) * rstd[k0] * gamma[k0] + beta[k0], 0.f);
    float a1 = fmaxf((xr[k1] - mu[k1]) * rstd[k1] * gamma[k1] + beta[k1], 0.f);
    v2f a = {a0, a1};
    v2f b = { W2[(size_t)k0 * COUT + n0 + lm],
              W2[(size_t)k1 * COUT + n0 + lm] };
    c = __builtin_amdgcn_wmma_f32_16x16x4_f32(false, a, false, b, (short)0, c, false, false);
  }
  float bias = b2[n0 + lm];
#pragma unroll
  for (int r = 0; r < 8; ++r) {
    int m = m0 + r + 8 * half;
    out[(size_t)m * COUT + n0 + lm] = c[r] + bias;
  }
}

// ---- launch ----
extern "C" void kernel_launch(void* const* d_in, const int* in_sizes, int n_in,
                              void* d_out, int out_size, void* d_ws, size_t ws_size,
                              hipStream_t stream) {
  (void)in_sizes; (void)n_in; (void)out_size; (void)ws_size;
  const float* x      = (const float*)d_in[0];
  const float* onehot = (const float*)d_in[1];
  const int*   adj    = (const int*)d_in[2];
  // d_in[3] = n_nodes (compile-time constant NN)
  const float* W1     = (const float*)d_in[4];
  const float* b1     = (const float*)d_in[5];
  const float* gamma  = (const float*)d_in[6];
  const float* beta   = (const float*)d_in[7];
  const float* W2     = (const float*)d_in[8];
  const float* b2     = (const float*)d_in[9];
  const float* cw1    = (const float*)d_in[10];
  const float* cb1    = (const float*)d_in[11];
  const float* cw2    = (const float*)d_in[12];
  const float* cb2    = (const float*)d_in[13];
  const float* Wr     = (const float*)d_in[14];
  const float* br     = (const float*)d_in[15];

  float* out_x2 = (float*)d_out;                       // [N, COUT]
  float* out_oh = out_x2 + (size_t)NN * COUT;          // [N, N] = new_oh

  float* ws      = (float*)d_ws;
  int*   deg     = (int*)ws;                           // NN
  int*   lists   = deg + NN;                           // NN * MAXDEG
  float* agg_x   = (float*)(lists + (size_t)NN * MAXDEG);  // NN*CIN
  float* readout = agg_x + (size_t)NN * CIN;           // NN*OHCH
  float* x2pre   = readout + (size_t)NN * OHCH;        // NN*COUT
  float* mu      = x2pre + (size_t)NN * COUT;          // COUT
  float* rstd    = mu + COUT;                          // COUT

  k_build_lists<<<NN, 256, 0, stream>>>(adj, deg, lists);
  k_gather_x<<<NN, CIN, 0, stream>>>(x, deg, lists, agg_x);
  k_gather_oh<<<NN, 256, 0, stream>>>(onehot, deg, lists, out_oh);
  k_sort_conv<<<NN, 1024, 0, stream>>>(out_oh, cw1, cb1, cw2, cb2, Wr, br, readout);
  k_gemm1<<<NN / 16, 256, 0, stream>>>(agg_x, readout, W1, b1, x2pre);
  k_bnstats<<<COUT, 256, 0, stream>>>(x2pre, mu, rstd);
  k_gemm2<<<NN / 16, 256, 0, stream>>>(x2pre, mu, rstd, gamma, beta, W2, b2, out_x2);
}